// BahdanauAttention_70652212019613
// MI455X (gfx1250) — compile-verified
//
#include <hip/hip_runtime.h>
#include <hip/hip_bf16.h>

// ---- problem constants (from reference) ----
#define B_  8
#define TE_ 512
#define TD_ 256
#define D_  256
#define U_  256

typedef __attribute__((ext_vector_type(2))) float v2f;
typedef __attribute__((ext_vector_type(4))) float v4f;
typedef __attribute__((ext_vector_type(8))) float v8f;

// Native CDNA5 transcendental tanh (v_tanh_f32) if the toolchain exposes it.
#if __has_builtin(__builtin_amdgcn_tanhf)
  #define FAST_TANH(x) __builtin_amdgcn_tanhf(x)
#elif __has_builtin(__builtin_amdgcn_tanh_f32)
  #define FAST_TANH(x) __builtin_amdgcn_tanh_f32(x)
#else
  #define FAST_TANH(x) tanhf(x)
#endif

// ---------------------------------------------------------------------------
// Kernel 1: out[M,256] = X[M,256] @ W[256,256] + bias   (fp32 WMMA 16x16x4)
// One wave per 16x16 output tile, 8 waves per 256-thread block.
// A layout (16x4 f32): lanes 0-15 -> K=0,1 ; lanes 16-31 -> K=2,3 (2 VGPRs)
// B layout (4x16 f32): VGPR v, lanes 0-15 -> row K=v ; lanes 16-31 -> K=v+2
// C/D layout: VGPR r, lanes 0-15 -> M=r ; lanes 16-31 -> M=r+8 ; N=lane%16
// ---------------------------------------------------------------------------
__global__ void __launch_bounds__(256) proj_wmma_f32(
    const float* __restrict__ X, const float* __restrict__ W,
    const float* __restrict__ bias, float* __restrict__ out)
{
  const int K = D_, U = U_;
  const int wave = threadIdx.x >> 5;
  const int lane = threadIdx.x & 31;
  const int tile = blockIdx.x * 8 + wave;
  const int tilesN = U / 16;
  const int tm = tile / tilesN;
  const int tn = tile % tilesN;

  const int half = lane >> 4;      // which lane-half
  const int lm   = lane & 15;
  const int arow = tm * 16 + lm;   // A-matrix row owned by this lane
  const int bcol = tn * 16 + lm;   // B/C/D column owned by this lane
  const int koff = half * 2;       // K sub-offset for this lane-half

  v8f acc = {};
  const float* __restrict__ arowp = X + (size_t)arow * K;
  for (int k = 0; k < K; k += 4) {
    v2f a, b;
    a.x = arowp[k + koff];
    a.y = arowp[k + koff + 1];
    b.x = W[(size_t)(k + koff)     * U + bcol];
    b.y = W[(size_t)(k + koff + 1) * U + bcol];
    acc = __builtin_amdgcn_wmma_f32_16x16x4_f32(
        /*neg_a=*/false, a, /*neg_b=*/false, b,
        /*c_mod=*/(short)0, acc, /*reuse_a=*/false, /*reuse_b=*/false);
  }

  const float bb = bias[bcol];
  const int rbase = tm * 16 + half * 8;
#pragma unroll
  for (int r = 0; r < 8; ++r)
    out[(size_t)(rbase + r) * U + bcol] = acc[r] + bb;
}

// ---------------------------------------------------------------------------
// Kernel 2: fused score + softmax. One block per (b,t):
//   logits[e] = sum_u tanh(ep[b,e,u] + dp[b,t,u]) * V[u] + bV
//   weights[b,t,e] = softmax_e(logits)
// Each lane owns u in {4*lane..4*lane+3, 128+4*lane..131+4*lane}; that set is
// constant over e, so dp[u] and V[u] are hoisted into 4 float4 registers.
// Hot loop per e: 2x global_load_b128 (coalesced 512B/wave) + 8x(add,tanh,fma)
// + 5 shuffles for the wave32 cross-lane sum.
// ---------------------------------------------------------------------------
__global__ void __launch_bounds__(256) score_softmax(
    const float* __restrict__ ep,   // [B*TE, U]
    const float* __restrict__ dp,   // [B*TD, U]
    const float* __restrict__ V,    // [U]
    const float* __restrict__ bV,   // [1]
    float* __restrict__ wts)        // [B*TD, TE]
{
  __shared__ float s_logit[TE_];
  __shared__ float s_red[256];

  const int bt   = blockIdx.x;      // b*TD + t
  const int b    = bt >> 8;         // TD_ == 256
  const int tid  = threadIdx.x;
  const int wave = tid >> 5;
  const int lane = tid & 31;

  // hoist the lane-private dp / V slices into registers (constant over e)
  const float* __restrict__ dprow = dp + (size_t)bt * U_;
  const v4f d0 = *(const v4f*)(dprow + 4 * lane);
  const v4f d1 = *(const v4f*)(dprow + 128 + 4 * lane);
  const v4f v0 = *(const v4f*)(V + 4 * lane);
  const v4f v1 = *(const v4f*)(V + 128 + 4 * lane);
  const float bv = bV[0];

  const float* __restrict__ epb = ep + (size_t)b * TE_ * U_;

  // each wave owns 64 consecutive encoder positions
  for (int i = 0; i < 64; ++i) {
    const int e = wave * 64 + i;
    const float* __restrict__ row = epb + (size_t)e * U_;
    const v4f r0 = *(const v4f*)(row + 4 * lane);
    const v4f r1 = *(const v4f*)(row + 128 + 4 * lane);
    float acc = 0.f;
#pragma unroll
    for (int c = 0; c < 4; ++c)
      acc += FAST_TANH(r0[c] + d0[c]) * v0[c];
#pragma unroll
    for (int c = 0; c < 4; ++c)
      acc += FAST_TANH(r1[c] + d1[c]) * v1[c];
#pragma unroll
    for (int off = 16; off >= 1; off >>= 1)
      acc += __shfl_xor(acc, off, 32);
    if (lane == 0) s_logit[e] = acc + bv;
  }
  __syncthreads();

  // ---- softmax over e (512 entries, 2 per thread) ----
  float m = fmaxf(s_logit[tid], s_logit[tid + 256]);
  s_red[tid] = m;
  __syncthreads();
  for (int s = 128; s > 0; s >>= 1) {
    if (tid < s) s_red[tid] = fmaxf(s_red[tid], s_red[tid + s]);
    __syncthreads();
  }
  m = s_red[0];
  __syncthreads();

  const float e0 = __expf(s_logit[tid]       - m);
  const float e1 = __expf(s_logit[tid + 256] - m);
  s_red[tid] = e0 + e1;
  __syncthreads();
  for (int s = 128; s > 0; s >>= 1) {
    if (tid < s) s_red[tid] += s_red[tid + s];
    __syncthreads();
  }
  const float inv = 1.f / s_red[0];

  wts[(size_t)bt * TE_ + tid]       = e0 * inv;
  wts[(size_t)bt * TE_ + tid + 256] = e1 * inv;
}

// ---------------------------------------------------------------------------
// Kernel 3: context[b,t,d] = sum_e wts[b,t,e] * enc[b,e,d]
// Batched GEMM via fp32 WMMA; one wave per 16x16 (t,d) tile, K=TE=512.
// ---------------------------------------------------------------------------
__global__ void __launch_bounds__(256) context_wmma_f32(
    const float* __restrict__ wts,  // [B,TD,TE]
    const float* __restrict__ enc,  // [B,TE,D]
    float* __restrict__ out)        // [B,TD,D]
{
  const int wave = threadIdx.x >> 5;
  const int lane = threadIdx.x & 31;
  const int tile = blockIdx.x * 8 + wave;
  const int b    = tile >> 8;       // (TD/16)*(D/16) = 256 tiles per batch
  const int rem  = tile & 255;
  const int tt   = rem >> 4;        // t-tile
  const int tdc  = rem & 15;        // d-tile

  const int half = lane >> 4;
  const int lm   = lane & 15;
  const int trow = tt * 16 + lm;
  const int dcol = tdc * 16 + lm;
  const int koff = half * 2;

  const float* __restrict__ wrow = wts + ((size_t)b * TD_ + trow) * TE_;
  const float* __restrict__ encb = enc + (size_t)b * TE_ * D_;

  v8f acc = {};
  for (int e = 0; e < TE_; e += 4) {
    v2f a, bm;
    a.x  = wrow[e + koff];
    a.y  = wrow[e + koff + 1];
    bm.x = encb[(size_t)(e + koff)     * D_ + dcol];
    bm.y = encb[(size_t)(e + koff + 1) * D_ + dcol];
    acc = __builtin_amdgcn_wmma_f32_16x16x4_f32(
        false, a, false, bm, (short)0, acc, false, false);
  }

  const int rbase = tt * 16 + half * 8;
#pragma unroll
  for (int r = 0; r < 8; ++r)
    out[((size_t)b * TD_ + rbase + r) * D_ + dcol] = acc[r];
}

// ---------------------------------------------------------------------------
extern "C" void kernel_launch(void* const* d_in, const int* in_sizes, int n_in,
                              void* d_out, int out_size, void* d_ws, size_t ws_size,
                              hipStream_t stream) {
  (void)in_sizes; (void)n_in; (void)out_size; (void)ws_size;

  const float* enc = (const float*)d_in[0];   // [B,TE,D]
  const float* dec = (const float*)d_in[1];   // [B,TD,D]
  const float* W1  = (const float*)d_in[2];   // [D,U]
  const float* b1  = (const float*)d_in[3];   // [U]
  const float* W2  = (const float*)d_in[4];   // [D,U]
  const float* b2  = (const float*)d_in[5];   // [U]
  const float* V   = (const float*)d_in[6];   // [U,1]
  const float* bV  = (const float*)d_in[7];   // [1]

  float* ctx = (float*)d_out;                           // [B,TD,D]
  float* wts = ctx + (size_t)B_ * TD_ * D_;             // [B,TD,TE] (output #2)

  float* ep = (float*)d_ws;                             // [B*TE,U]  (4 MB)
  float* dq = ep + (size_t)B_ * TE_ * U_;               // [B*TD,U]  (2 MB)

  // enc_proj: 4096 wave-tiles -> 512 blocks of 8 waves
  proj_wmma_f32<<<(B_ * TE_ / 16) * (U_ / 16) / 8, 256, 0, stream>>>(enc, W1, b1, ep);
  // dec_proj: 2048 wave-tiles -> 256 blocks
  proj_wmma_f32<<<(B_ * TD_ / 16) * (U_ / 16) / 8, 256, 0, stream>>>(dec, W2, b2, dq);
  // fused score + softmax: one block per (b,t)
  score_softmax<<<B_ * TD_, 256, 0, stream>>>(ep, dq, V, bV, wts);
  // context: 8 batches * 256 tiles -> 256 blocks
  context_wmma_f32<<<B_ * (TD_ / 16) * (D_ / 16) / 8, 256, 0, stream>>>(wts, enc, ctx);
}